// ModelRank_12549894439330
// MI455X (gfx1250) — compile-verified
//
#include <hip/hip_runtime.h>
#include <hip/hip_bf16.h>
#include <math.h>

#define LDIM 384
#define EDG  12288
#define KSEL 192
#define VOXD 640
#define NPIX (LDIM*LDIM)     /* 147456 */
#define NTOK (KSEL*KSEL)     /* 36864  */
#define DMOD 128
#define PADW 386             /* 384 + 2 halo for 3x3 conv */

#define CDIV(a,b) (((a)+(b)-1)/(b))

typedef __attribute__((ext_vector_type(2))) float v2f;
typedef __attribute__((ext_vector_type(8))) float v8f;

__device__ __forceinline__ float gelu_f(float x){ return 0.5f*x*(1.f+erff(x*0.70710678118f)); }
__device__ __forceinline__ float act_f(float v, int act){
  if (act==1) return fmaxf(v,0.f);
  if (act==2) return gelu_f(v);
  return v;
}
__device__ __forceinline__ void atomicMaxF(float* addr, float val){
  unsigned int* ai = (unsigned int*)addr;
  unsigned int old = *ai;
  while (true) {
    float fold = __uint_as_float(old);
    if (fold >= val) break;
    unsigned int assumed = old;
    old = atomicCAS(ai, assumed, __float_as_uint(val));
    if (old == assumed) break;
  }
}

#define WMMA_F32(a,b,c) __builtin_amdgcn_wmma_f32_16x16x4_f32(false,(a),false,(b),(short)0,(c),false,false)

// ---------------------------------------------------------------------------
// fp32 WMMA GEMM: C[M,N] = act(A[M,K] @ B[K,N] + bias[N])
// Requirements: M % 16 == 0, N % 32 == 0.  K arbitrary (ragged tail handled
// out of the hot loop).  One 16x32 tile per wave: 2 accumulators sharing the
// A fragment; main loop is branch-free (no exec toggling around loads).
// fp32 16x16x4 fragment layout per CDNA5 ISA:
//   A: lanes 0-15 -> rows, VGPR0=K+0/VGPR1=K+1 ; lanes 16-31 -> K+2/K+3
//   B: mirrored;  C: VGPR j: lanes 0-15 -> M=j, lanes 16-31 -> M=8+j
// ---------------------------------------------------------------------------
__global__ void wmma_gemm_f32(const float* __restrict__ A, const float* __restrict__ B,
                              const float* __restrict__ bias, float* __restrict__ C,
                              int M, int N, int Kd, int act) {
  const int wave = threadIdx.x >> 5;
  const int lane = threadIdx.x & 31;
  const int tilesN = N >> 5;
  const int tile = blockIdx.x * (blockDim.x >> 5) + wave;
  if (tile >= (M >> 4) * tilesN) return;        // wave-uniform exit
  const int tm = tile / tilesN, tn = tile % tilesN;
  const int lo = lane & 15;
  const int hi = lane >> 4;
  const int rowA = tm*16 + lo;
  const int col0 = tn*32 + lo;
  const int col1 = col0 + 16;
  const float* __restrict__ Ap = A + (long)rowA*Kd + hi*2;
  const float* __restrict__ Bh = B + (long)(hi*2)*N;
  v8f acc0 = {}, acc1 = {};
  const int Kmain = Kd & ~3;
  for (int kk = 0; kk < Kmain; kk += 4) {
    v2f a, b0, b1;
    a.x = Ap[kk];
    a.y = Ap[kk+1];
    const float* Bk = Bh + (long)kk*N;
    b0.x = Bk[col0];     b0.y = Bk[N+col0];
    b1.x = Bk[col1];     b1.y = Bk[N+col1];
    __builtin_prefetch(Ap + kk + 32, 0, 1);
    acc0 = WMMA_F32(a, b0, acc0);
    acc1 = WMMA_F32(a, b1, acc1);
  }
  if (Kmain < Kd) {                              // ragged-K tail (uniform cond)
    const int k0 = Kmain + hi*2;
    v2f a, b0, b1;
    a.x = (k0   < Kd) ? Ap[Kmain]   : 0.f;
    a.y = (k0+1 < Kd) ? Ap[Kmain+1] : 0.f;
    b0.x = (k0   < Kd) ? B[(long)k0    *N + col0] : 0.f;
    b0.y = (k0+1 < Kd) ? B[(long)(k0+1)*N + col0] : 0.f;
    b1.x = (k0   < Kd) ? B[(long)k0    *N + col1] : 0.f;
    b1.y = (k0+1 < Kd) ? B[(long)(k0+1)*N + col1] : 0.f;
    acc0 = WMMA_F32(a, b0, acc0);
    acc1 = WMMA_F32(a, b1, acc1);
  }
  const int rbase = tm*16 + hi*8;
#pragma unroll
  for (int j = 0; j < 8; ++j) {
    long r = rbase + j;
    C[r*N + col0] = act_f(acc0[j] + bias[col0], act);
    C[r*N + col1] = act_f(acc1[j] + bias[col1], act);
  }
}

// ---------------------------------------------------------------------------
// Fused pair-stack GEMM (c2d2, M=NPIX, N=128, K=256).  A operand is the
// 256-channel pair tensor generated on the fly from pre-normalized sources:
//   k<64   : n1[h][k]      (temp1, IN+GELU applied with gains g[0:64])
//   k<128  : n2[w][k-64]   (temp2, gains g[64:128])
//   k>=128 : ocN[p][k-128] (c2d1 output, IN+GELU in place)
// Three branch-free K-phase loops; never materializes the 151 MB tensor.
// ---------------------------------------------------------------------------
__global__ void wmma_gemm_pair_f32(const float* __restrict__ n1, const float* __restrict__ n2,
                                   const float* __restrict__ ocN,
                                   const float* __restrict__ B, const float* __restrict__ bias,
                                   float* __restrict__ C) {
  const int wave = threadIdx.x >> 5;
  const int lane = threadIdx.x & 31;
  const int tile = blockIdx.x * (blockDim.x >> 5) + wave;
  if (tile >= (NPIX >> 4) * 4) return;
  const int tm = tile >> 2, tn = tile & 3;
  const int lo = lane & 15;
  const int hi = lane >> 4;
  const int p  = tm*16 + lo;
  const int col0 = tn*32 + lo;
  const int col1 = col0 + 16;
  const float* __restrict__ a1 = n1  + (p/LDIM)*64  + hi*2;
  const float* __restrict__ a2 = n2  + (p%LDIM)*64  + hi*2;
  const float* __restrict__ a3 = ocN + (long)p*128  + hi*2;
  const float* __restrict__ Bh = B + (long)(hi*2)*128;
  v8f acc0 = {}, acc1 = {};
  for (int kk = 0; kk < 64; kk += 4) {
    v2f a, b0, b1;
    a.x = a1[kk]; a.y = a1[kk+1];
    const float* Bk = Bh + (long)kk*128;
    b0.x = Bk[col0]; b0.y = Bk[128+col0];
    b1.x = Bk[col1]; b1.y = Bk[128+col1];
    acc0 = WMMA_F32(a, b0, acc0);
    acc1 = WMMA_F32(a, b1, acc1);
  }
  for (int kk = 0; kk < 64; kk += 4) {
    v2f a, b0, b1;
    a.x = a2[kk]; a.y = a2[kk+1];
    const float* Bk = Bh + (long)(64+kk)*128;
    b0.x = Bk[col0]; b0.y = Bk[128+col0];
    b1.x = Bk[col1]; b1.y = Bk[128+col1];
    acc0 = WMMA_F32(a, b0, acc0);
    acc1 = WMMA_F32(a, b1, acc1);
  }
  for (int kk = 0; kk < 128; kk += 4) {
    v2f a, b0, b1;
    a.x = a3[kk]; a.y = a3[kk+1];
    const float* Bk = Bh + (long)(128+kk)*128;
    b0.x = Bk[col0]; b0.y = Bk[128+col0];
    b1.x = Bk[col1]; b1.y = Bk[128+col1];
    __builtin_prefetch(a3 + kk + 32, 0, 1);
    acc0 = WMMA_F32(a, b0, acc0);
    acc1 = WMMA_F32(a, b1, acc1);
  }
  const int rbase = tm*16 + hi*8;
#pragma unroll
  for (int j = 0; j < 8; ++j) {
    long r = rbase + j;
    C[r*128 + col0] = acc0[j] + bias[col0];
    C[r*128 + col1] = acc1[j] + bias[col1];
  }
}

// ---------------------------------------------------------------------------
// WMMA 3x3 stride-2 conv, 128->128, via 9 accumulated GEMM taps.
// Input is zero-padded (PADW x PADW x 128); weights pre-transposed to
// Bt[tap][ci][d].  Output rows of a 16-row tile share ho (192 % 16 == 0),
// inner loops are branch-free; 576 WMMAs per wave, one C write.
// ---------------------------------------------------------------------------
__global__ void wmma_conv_f32(const float* __restrict__ pairP, const float* __restrict__ Bt,
                              const float* __restrict__ bias, float* __restrict__ outp) {
  const int wave = threadIdx.x >> 5;
  const int lane = threadIdx.x & 31;
  const int tile = blockIdx.x * (blockDim.x >> 5) + wave;
  if (tile >= (NTOK >> 4) * 4) return;
  const int tm = tile >> 2, tn = tile & 3;
  const int lo = lane & 15;
  const int hi = lane >> 4;
  const int po = tm*16 + lo;
  const int ho = po / KSEL, wo = po % KSEL;
  const int col0 = tn*32 + lo;
  const int col1 = col0 + 16;
  v8f acc0, acc1;
  const float b0v = bias[col0], b1v = bias[col1];
#pragma unroll
  for (int j = 0; j < 8; ++j) { acc0[j] = b0v; acc1[j] = b1v; }
  for (int tap = 0; tap < 9; ++tap) {
    const int ky = tap / 3, kx = tap % 3;
    const float* __restrict__ Ap = pairP + ((long)(2*ho + ky)*PADW + (2*wo + kx))*128 + hi*2;
    const float* __restrict__ Bh = Bt + (long)tap*128*128 + (long)(hi*2)*128;
    for (int kk = 0; kk < 128; kk += 4) {
      v2f a, b0, b1;
      a.x = Ap[kk]; a.y = Ap[kk+1];
      const float* Bk = Bh + (long)kk*128;
      b0.x = Bk[col0]; b0.y = Bk[128+col0];
      b1.x = Bk[col1]; b1.y = Bk[128+col1];
      __builtin_prefetch(Ap + kk + 32, 0, 1);
      acc0 = WMMA_F32(a, b0, acc0);
      acc1 = WMMA_F32(a, b1, acc1);
    }
  }
  const int rbase = tm*16 + hi*8;
#pragma unroll
  for (int j = 0; j < 8; ++j) {
    long r = rbase + j;
    outp[r*128 + col0] = acc0[j];
    outp[r*128 + col1] = acc1[j];
  }
}

// ----------------------------- small kernels -------------------------------
__global__ void k_fillf(float* p, float v, long n) {
  long i = blockIdx.x*(long)blockDim.x + threadIdx.x;
  if (i < n) p[i] = v;
}

__global__ void k_voro(const float* __restrict__ vf, const float* __restrict__ vn,
                       const float* __restrict__ W, const float* __restrict__ b,
                       const float* __restrict__ g, const float* __restrict__ bt,
                       float* __restrict__ voro) {
  int i = blockIdx.x*blockDim.x + threadIdx.x;
  if (i >= LDIM) return;
  float x[5] = { vf[i], vf[LDIM+i], vn[i], vn[LDIM+i], vn[2*LDIM+i] };
  float h[64];
  float mean = 0.f;
  for (int c=0;c<64;++c){
    float s = b[c];
    for (int k=0;k<5;++k) s += x[k]*W[k*64+c];
    s = fmaxf(s,0.f);
    h[c]=s; mean += s;
  }
  mean *= (1.f/64.f);
  float var=0.f;
  for (int c=0;c<64;++c){ float d=h[c]-mean; var += d*d; }
  var *= (1.f/64.f);
  float inv = rsqrtf(var+1e-5f);
  for (int c=0;c<64;++c) voro[i*64+c] = (h[c]-mean)*inv*g[c] + bt[c];
}

__global__ void k_node(const float* __restrict__ f1d, const float* __restrict__ feat,
                       const float* __restrict__ voro, const float* __restrict__ mpnn,
                       float* __restrict__ node) {
  int idx = blockIdx.x*blockDim.x + threadIdx.x;
  if (idx >= LDIM*193) return;
  int i = idx / 193, c = idx % 193;
  float v;
  if (c < 74)       v = f1d[c*LDIM + i];
  else if (c < 97)  v = feat[i*23 + (c-74)];
  else if (c < 161) v = voro[i*64 + (c-97)];
  else              v = mpnn[i*32 + (c-161)];
  node[idx] = v;
}

__global__ void k_agg(const float* __restrict__ node, const int* __restrict__ adj,
                      float* __restrict__ agg) {
  long idx = blockIdx.x*(long)blockDim.x + threadIdx.x;
  if (idx >= (long)EDG*193) return;
  int e = idx / 193, c = idx % 193;
  int s = adj[e*2+0], d = adj[e*2+1];
  atomicAdd(&agg[d*193+c], node[s*193+c]);
}

__global__ void k_score(const float* __restrict__ agg, const float* __restrict__ node,
                        const float* __restrict__ Wrel, const float* __restrict__ brel,
                        const float* __restrict__ Wroot, float* __restrict__ score) {
  int i = blockIdx.x*blockDim.x + threadIdx.x;
  if (i >= LDIM) return;
  float s = brel[0];
  for (int c=0;c<193;++c) s += agg[i*193+c]*Wrel[c] + node[i*193+c]*Wroot[c];
  score[i] = tanhf(s);
}

__global__ void k_topk(const float* __restrict__ score, int* __restrict__ perm,
                       float* __restrict__ vals, int* __restrict__ inv) {
  __shared__ float s[LDIM];
  int i = threadIdx.x;
  if (i < LDIM) s[i] = score[i];
  __syncthreads();
  if (i < LDIM) {
    float mine = s[i];
    int rank = 0;
    for (int j = 0; j < LDIM; ++j) {
      float o = s[j];
      if (o > mine || (o == mine && j < i)) ++rank;
    }
    if (rank < KSEL) { perm[rank] = i; vals[rank] = mine; inv[i] = rank; }
    else inv[i] = -1;
  }
}

__global__ void k_nodep(const float* __restrict__ node, const int* __restrict__ perm,
                        const float* __restrict__ vals, float* __restrict__ node_p) {
  int idx = blockIdx.x*blockDim.x + threadIdx.x;
  if (idx >= KSEL*193) return;
  int r = idx / 193, c = idx % 193;
  node_p[idx] = node[perm[r]*193 + c] * vals[r];
}

__global__ void k_remap(const int* __restrict__ adj, const int* __restrict__ inv,
                        int* __restrict__ ns, int* __restrict__ nd, int* __restrict__ em) {
  int e = blockIdx.x*blockDim.x + threadIdx.x;
  if (e >= EDG) return;
  int a = inv[adj[e*2+0]], b = inv[adj[e*2+1]];
  em[e] = (a >= 0 && b >= 0) ? 1 : 0;
  ns[e] = a < 0 ? 0 : a;
  nd[e] = b < 0 ? 0 : b;
}

__global__ void k_gat_logits(const float* __restrict__ xl, const float* __restrict__ xr,
                             const float* __restrict__ att, const int* __restrict__ ns,
                             const int* __restrict__ nd, const int* __restrict__ em,
                             float* __restrict__ logits, int Hh, int Cc) {
  int idx = blockIdx.x*blockDim.x + threadIdx.x;
  if (idx >= EDG*Hh) return;
  int e = idx / Hh, hh = idx % Hh;
  if (!em[e]) { logits[idx] = -1e30f; return; }
  int HC = Hh*Cc;
  const float* l = xl + (long)ns[e]*HC + hh*Cc;
  const float* r = xr + (long)nd[e]*HC + hh*Cc;
  const float* a = att + hh*Cc;
  float s = 0.f;
  for (int c=0;c<Cc;++c) {
    float v = l[c] + r[c];
    v = v > 0.f ? v : 0.2f*v;
    s += v*a[c];
  }
  logits[idx] = s;
}

__global__ void k_gat_max(const float* __restrict__ logits, const int* __restrict__ nd,
                          float* __restrict__ mbuf, int Hh) {
  int idx = blockIdx.x*blockDim.x + threadIdx.x;
  if (idx >= EDG*Hh) return;
  int e = idx / Hh, hh = idx % Hh;
  atomicMaxF(&mbuf[nd[e]*Hh + hh], logits[idx]);
}

__global__ void k_gat_fixm(float* __restrict__ mbuf, int n) {
  int i = blockIdx.x*blockDim.x + threadIdx.x;
  if (i < n && mbuf[i] < -1e29f) mbuf[i] = 0.f;
}

__global__ void k_gat_wden(const float* __restrict__ logits, const int* __restrict__ nd,
                           const float* __restrict__ mbuf, float* __restrict__ wbuf,
                           float* __restrict__ den, int Hh) {
  int idx = blockIdx.x*blockDim.x + threadIdx.x;
  if (idx >= EDG*Hh) return;
  int e = idx / Hh, hh = idx % Hh;
  float w = expf(logits[idx] - mbuf[nd[e]*Hh + hh]);
  wbuf[idx] = w;
  atomicAdd(&den[nd[e]*Hh + hh], w);
}

__global__ void k_gat_out(const float* __restrict__ wbuf, const float* __restrict__ den,
                          const float* __restrict__ xl, const int* __restrict__ ns,
                          const int* __restrict__ nd, float* __restrict__ outb,
                          int Hh, int Cc) {
  long idx = blockIdx.x*(long)blockDim.x + threadIdx.x;
  int HC = Hh*Cc;
  if (idx >= (long)EDG*HC) return;
  int e = idx / HC, rem = idx % HC;
  int hh = rem / Cc, c = rem % Cc;
  float alpha = wbuf[e*Hh+hh] / (den[nd[e]*Hh+hh] + 1e-16f);
  atomicAdd(&outb[nd[e]*HC + hh*Cc + c], alpha * xl[(long)ns[e]*HC + hh*Cc + c]);
}

__global__ void k_bias_act(float* __restrict__ x, const float* __restrict__ bias,
                           int n, int C, int act) {
  int i = blockIdx.x*blockDim.x + threadIdx.x;
  if (i >= n) return;
  x[i] = act_f(x[i] + bias[i % C], act);
}

__global__ void k_xnode(const float* __restrict__ h, const float* __restrict__ w,
                        const float* __restrict__ b, float* __restrict__ xnode) {
  int i = threadIdx.x;
  if (i >= KSEL) return;
  float s = b[0];
  for (int c=0;c<64;++c) s += h[i*64+c]*w[c];
  xnode[i] = s;
}

__global__ void k_scatter(const float* __restrict__ val, const int* __restrict__ vidx,
                          float* __restrict__ f3dA, int n) {
  int i = blockIdx.x*blockDim.x + threadIdx.x;
  if (i >= n) return;
  atomicAdd(&f3dA[vidx[i]], val[i]);
}

__global__ void k_fused_cat(const float* __restrict__ t2d, const float* __restrict__ profile,
                            const float* __restrict__ entropy, const float* __restrict__ afor,
                            const float* __restrict__ mask3d,
                            const float* __restrict__ w1, const float* __restrict__ b1,
                            const float* __restrict__ w2, const float* __restrict__ b2,
                            float* __restrict__ cat48) {
  int p = blockIdx.x*blockDim.x + threadIdx.x;
  if (p >= NPIX) return;
  float xin[43];
  for (int j=0;j<36;++j) xin[j] = profile[(long)p*36+j];
  xin[36] = entropy[p];
  for (int j=0;j<6;++j) xin[37+j] = afor[(long)p*6+j];
  float hid[40];
  for (int j=0;j<40;++j){
    float s = b1[j];
    for (int k=0;k<43;++k) s += xin[k]*w1[k*40+j];
    hid[j] = fmaxf(s,0.f);
  }
  float m = mask3d[p];
  float* dst = cat48 + (long)p*48;
  for (int c=0;c<32;++c) dst[c] = t2d[(long)p*32+c];
  for (int j=0;j<16;++j){
    float s = b2[j];
    for (int k=0;k<40;++k) s += hid[k]*w2[k*16+j];
    dst[32+j] = s*m;
  }
}

__global__ void k_chan_stats(const float* __restrict__ src, float* __restrict__ mean,
                             float* __restrict__ var, long nPix, int C) {
  int c = blockIdx.x;
  __shared__ float sm[256], sv[256];
  float s=0.f, ss=0.f;
  for (long p = threadIdx.x; p < nPix; p += blockDim.x) {
    float v = src[p*C + c];
    s += v; ss += v*v;
  }
  sm[threadIdx.x]=s; sv[threadIdx.x]=ss;
  __syncthreads();
  for (int st=128; st>0; st>>=1){
    if (threadIdx.x<st){ sm[threadIdx.x]+=sm[threadIdx.x+st]; sv[threadIdx.x]+=sv[threadIdx.x+st]; }
    __syncthreads();
  }
  if (threadIdx.x==0){
    float m = sm[0]/(float)nPix;
    mean[c]=m; var[c]=sv[0]/(float)nPix - m*m;
  }
}

// normalize nf3d once for the two broadcast channel blocks (gains 0:64, 64:128)
__global__ void k_nf3d_norm(const float* __restrict__ nf3d, const float* __restrict__ m1,
                            const float* __restrict__ v1, const float* __restrict__ g,
                            const float* __restrict__ b, float* __restrict__ n1,
                            float* __restrict__ n2) {
  int idx = blockIdx.x*blockDim.x + threadIdx.x;
  if (idx >= LDIM*64) return;
  int c = idx & 63;
  float x = nf3d[idx];
  float inv = rsqrtf(v1[c]+1e-6f);
  float xm = (x - m1[c]) * inv;
  n1[idx] = gelu_f(xm*g[c]    + b[c]);
  n2[idx] = gelu_f(xm*g[64+c] + b[64+c]);
}

__global__ void k_inorm_gelu(float* __restrict__ x, const float* __restrict__ mean,
                             const float* __restrict__ var, const float* __restrict__ g,
                             const float* __restrict__ b, long nElem, int C) {
  long idx = blockIdx.x*(long)blockDim.x + threadIdx.x;
  if (idx >= nElem) return;
  int c = idx % C;
  float y = (x[idx]-mean[c])*rsqrtf(var[c]+1e-6f)*g[c]+b[c];
  x[idx] = gelu_f(y);
}

// instance-norm + gelu, writing into zero-padded conv input layout
__global__ void k_inorm_pad(const float* __restrict__ x, const float* __restrict__ mean,
                            const float* __restrict__ var, const float* __restrict__ g,
                            const float* __restrict__ b, float* __restrict__ outp) {
  long idx = blockIdx.x*(long)blockDim.x + threadIdx.x;
  if (idx >= (long)NPIX*128) return;
  int c = idx % 128;
  long p = idx / 128;
  int w = p % LDIM, h = p / LDIM;
  float y = (x[idx]-mean[c])*rsqrtf(var[c]+1e-6f)*g[c]+b[c];
  outp[((long)(h+1)*PADW + (w+1))*128 + c] = gelu_f(y);
}

// samp_w (OIHW 128x128x3x3) -> Bt[tap][ci][d]
__global__ void k_transw(const float* __restrict__ w, float* __restrict__ Bt) {
  int idx = blockIdx.x*blockDim.x + threadIdx.x;
  if (idx >= 9*128*128) return;
  int d = idx % 128;
  int rest = idx / 128;
  int ci = rest % 128;
  int tap = rest / 128;
  Bt[idx] = w[(long)d*1152 + ci*9 + tap];
}

__global__ void k_ln(const float* __restrict__ x, const float* __restrict__ g,
                     const float* __restrict__ b, float* __restrict__ y, int nTok) {
  int t = blockIdx.x*blockDim.x + threadIdx.x;
  if (t >= nTok) return;
  const float* xp = x + (long)t*DMOD;
  float m=0.f;
  for (int c=0;c<DMOD;++c) m += xp[c];
  m *= (1.f/DMOD);
  float v=0.f;
  for (int c=0;c<DMOD;++c){ float d0=xp[c]-m; v += d0*d0; }
  v *= (1.f/DMOD);
  float inv = rsqrtf(v + 1e-5f);
  float* yp = y + (long)t*DMOD;
  for (int c=0;c<DMOD;++c) yp[c] = (xp[c]-m)*inv*g[c] + b[c];
}

// online-softmax attention, one thread per (batch, head, query)
__global__ void k_attn(const float* __restrict__ qb, const float* __restrict__ kb,
                       const float* __restrict__ vb, float* __restrict__ ob, int colmode) {
  int idx = blockIdx.x*blockDim.x + threadIdx.x;
  const int S = KSEL, Hh = 4, dh = 32;
  if (idx >= S*Hh*S) return;
  int b = idx / (Hh*S);
  int rem = idx % (Hh*S);
  int hh = rem / S;
  int q = rem % S;
  long tq = colmode ? ((long)q*S + b) : ((long)b*S + q);
  const float* qv = qb + tq*DMOD + hh*dh;
  float mx = -1e30f, l = 0.f;
  float o[32];
#pragma unroll
  for (int c=0;c<32;++c) o[c]=0.f;
  const float scale = 0.17677669529f; // 1/sqrt(32)
  for (int s2=0;s2<S;++s2) {
    long tk = colmode ? ((long)s2*S + b) : ((long)b*S + s2);
    const float* kv = kb + tk*DMOD + hh*dh;
    float lg=0.f;
#pragma unroll
    for (int c=0;c<dh;++c) lg += qv[c]*kv[c];
    lg *= scale;
    float nm = fmaxf(mx, lg);
    float corr = expf(mx-nm);
    float w = expf(lg-nm);
    l = l*corr + w;
    const float* vv = vb + tk*DMOD + hh*dh;
#pragma unroll
    for (int c=0;c<dh;++c) o[c] = o[c]*corr + w*vv[c];
    mx = nm;
  }
  float inv_l = 1.f/l;
  float* op = ob + tq*DMOD + hh*dh;
#pragma unroll
  for (int c=0;c<dh;++c) op[c] = o[c]*inv_l;
}

__global__ void k_add(float* __restrict__ dst, const float* __restrict__ src, long n) {
  long i = blockIdx.x*(long)blockDim.x + threadIdx.x;
  if (i < n) dst[i] += src[i];
}

__global__ void k_maskdot(const float* __restrict__ x2, const float* __restrict__ w,
                          const float* __restrict__ b, float* __restrict__ maskv) {
  int p = blockIdx.x*blockDim.x + threadIdx.x;
  if (p >= NTOK) return;
  float s = b[0];
  const float* xp = x2 + (long)p*DMOD;
  for (int c=0;c<DMOD;++c) s += xp[c]*w[c];
  maskv[p] = s;
}

__global__ void k_final(const float* __restrict__ xnode, const float* __restrict__ maskv,
                        float* __restrict__ out) {
  __shared__ float sm[256];
  float s = 0.f;
  for (int p = threadIdx.x; p < NTOK; p += 256)
    s += xnode[p/KSEL] * maskv[p];
  sm[threadIdx.x] = s;
  __syncthreads();
  for (int st=128; st>0; st>>=1){
    if (threadIdx.x<st) sm[threadIdx.x]+=sm[threadIdx.x+st];
    __syncthreads();
  }
  if (threadIdx.x==0) out[0] = sm[0] / (float)NTOK;
}

// ------------------------------- host side ---------------------------------
static void gemm(hipStream_t st, const float* A, const float* B, const float* bias,
                 float* C, int M, int N, int Kd, int act) {
  int tiles = (M >> 4) * (N >> 5);      // M%16==0, N%32==0 at every call site
  wmma_gemm_f32<<<CDIV(tiles,8), 256, 0, st>>>(A, B, bias, C, M, N, Kd, act);
}

static void run_gat(hipStream_t st, const float* x, int fin, int Hh, int Cc,
                    const float* Wl, const float* bl, const float* Wr, const float* br,
                    const float* att, const float* bias,
                    const int* ns, const int* nd, const int* em,
                    float* xl, float* xr, float* logits, float* wbuf,
                    float* mbuf, float* den, float* outb, int act) {
  int HC = Hh*Cc;
  gemm(st, x, Wl, bl, xl, KSEL, HC, fin, 0);
  gemm(st, x, Wr, br, xr, KSEL, HC, fin, 0);
  k_gat_logits<<<CDIV(EDG*Hh,256),256,0,st>>>(xl,xr,att,ns,nd,em,logits,Hh,Cc);
  k_fillf<<<CDIV(KSEL*Hh,256),256,0,st>>>(mbuf,-1e30f,(long)KSEL*Hh);
  k_gat_max<<<CDIV(EDG*Hh,256),256,0,st>>>(logits,nd,mbuf,Hh);
  k_gat_fixm<<<CDIV(KSEL*Hh,256),256,0,st>>>(mbuf,KSEL*Hh);
  k_fillf<<<CDIV(KSEL*Hh,256),256,0,st>>>(den,0.f,(long)KSEL*Hh);
  k_gat_wden<<<CDIV(EDG*Hh,256),256,0,st>>>(logits,nd,mbuf,wbuf,den,Hh);
  k_fillf<<<CDIV(KSEL*HC,256),256,0,st>>>(outb,0.f,(long)KSEL*HC);
  k_gat_out<<<CDIV(EDG*HC,256),256,0,st>>>(wbuf,den,xl,ns,nd,outb,Hh,Cc);
  k_bias_act<<<CDIV(KSEL*HC,256),256,0,st>>>(outb,bias,KSEL*HC,HC,act);
}

// param/input indices per setup_inputs() dict order, depth-first
enum {
  IP_VORO_W=0, IP_VORO_B, IP_VORO_G, IP_VORO_BT,
  IP_SAG_WREL, IP_SAG_BREL, IP_SAG_WROOT,
  IP_GAT0=7, IP_GAT1=13, IP_GAT2=19, IP_GAT3=25,   // Wl,bl,Wr,br,att,bias
  IP_LIN1_W=31, IP_LIN1_B=32,
  IP_C1D_W=33, IP_C1D_B=34,
  IP_FUS_W1=35, IP_FUS_B1=36, IP_FUS_W2=37, IP_FUS_B2=38,
  IP_C2D1_W=39, IP_C2D1_B=40,
  IP_IN1_G=41, IP_IN1_B=42,
  IP_C2D2_W=43, IP_C2D2_B=44,
  IP_IN2_G=45, IP_IN2_B=46,
  IP_SAMP_W=47, IP_SAMP_B=48,
  IP_LN1_G=49, IP_LN1_B=50,
  IP_ROW=51,                                        // Wq,bq,Wk,bk,Wv,bv,Wo,bo
  IP_LN2_G=59, IP_LN2_B=60,
  IP_COL=61,
  IP_LN3_G=69, IP_LN3_B=70,
  IP_FF_W1=71, IP_FF_B1=72, IP_FF_W2=73, IP_FF_B2=74,
  IP_MASK_W=75, IP_MASK_B=76,
  IN_VORO_FEAT=77, IN_VORO_NORM=78, IN_F1D=79, IN_FEAT=80, IN_MPNN=81, IN_VAL=82,
  IN_2D=83, IN_PROFILE=84, IN_ENTROPY=85, IN_MASK3D=86, IN_AFOR=87,
  IN_ADJ=88, IN_VIDX=89
};

extern "C" void kernel_launch(void* const* d_in, const int* in_sizes, int n_in,
                              void* d_out, int out_size, void* d_ws, size_t ws_size,
                              hipStream_t stream) {
  (void)in_sizes; (void)n_in; (void)out_size; (void)ws_size;
  auto P  = [&](int i)->const float*{ return (const float*)d_in[i]; };
  auto PI = [&](int i)->const int*  { return (const int*)d_in[i]; };

  char* wsp = (char*)d_ws;
  size_t off = 0;
  auto allocF = [&](size_t n)->float*{
    float* p=(float*)(wsp+off); off += n*sizeof(float); off=(off+255)&~((size_t)255); return p; };
  auto allocI = [&](size_t n)->int*{
    int* p=(int*)(wsp+off); off += n*sizeof(int); off=(off+255)&~((size_t)255); return p; };

  float* voro   = allocF((size_t)LDIM*64);
  float* node   = allocF((size_t)LDIM*193);
  float* agg    = allocF((size_t)LDIM*193);
  float* score  = allocF(LDIM);
  float* vals   = allocF(KSEL);
  float* node_p = allocF((size_t)KSEL*193);
  float* xl     = allocF((size_t)KSEL*128);
  float* xr     = allocF((size_t)KSEL*128);
  float* logits = allocF((size_t)EDG*8);
  float* wbuf   = allocF((size_t)EDG*8);
  float* mbuf   = allocF((size_t)KSEL*8);
  float* den    = allocF((size_t)KSEL*8);
  float* hbuf0  = allocF((size_t)KSEL*128);
  float* hbuf1  = allocF((size_t)KSEL*128);
  float* xnode  = allocF(KSEL);
  float* f3dA   = allocF((size_t)LDIM*VOXD);
  float* nf3d   = allocF((size_t)LDIM*64);
  float* nf3dN1 = allocF((size_t)LDIM*64);
  float* nf3dN2 = allocF((size_t)LDIM*64);
  float* m1     = allocF(64);
  float* v1     = allocF(64);
  float* moc    = allocF(128);
  float* voc    = allocF(128);
  float* m2     = allocF(128);
  float* v2     = allocF(128);
  float* maskv  = allocF(NTOK);
  float* Bt     = allocF((size_t)9*128*128);
  int*   perm   = allocI(KSEL);
  int*   inv    = allocI(LDIM);
  int*   ns     = allocI(EDG);
  int*   nd     = allocI(EDG);
  int*   em     = allocI(EDG);
  float* cat48  = allocF((size_t)NPIX*48);
  float* oc     = allocF((size_t)NPIX*128);
  float* pairB  = allocF((size_t)NPIX*128);
  float* pairP  = allocF((size_t)PADW*PADW*128);
  float* x2     = allocF((size_t)NTOK*DMOD);
  float* xln    = allocF((size_t)NTOK*DMOD);
  float* qb     = allocF((size_t)NTOK*DMOD);
  float* kb     = allocF((size_t)NTOK*DMOD);
  float* vb     = allocF((size_t)NTOK*DMOD);
  float* ab     = allocF((size_t)NTOK*DMOD);
  float* tmp    = allocF((size_t)NTOK*DMOD);
  float* ffh    = allocF((size_t)NTOK*256);

  // ---- node features -------------------------------------------------------
  k_voro<<<CDIV(LDIM,128),128,0,stream>>>(P(IN_VORO_FEAT), P(IN_VORO_NORM),
      P(IP_VORO_W), P(IP_VORO_B), P(IP_VORO_G), P(IP_VORO_BT), voro);
  k_node<<<CDIV(LDIM*193,256),256,0,stream>>>(P(IN_F1D), P(IN_FEAT), voro, P(IN_MPNN), node);

  // ---- SAG pool ------------------------------------------------------------
  k_fillf<<<CDIV(LDIM*193,256),256,0,stream>>>(agg, 0.f, (long)LDIM*193);
  k_agg<<<CDIV(EDG*193,256),256,0,stream>>>(node, PI(IN_ADJ), agg);
  k_score<<<CDIV(LDIM,128),128,0,stream>>>(agg, node, P(IP_SAG_WREL), P(IP_SAG_BREL),
                                           P(IP_SAG_WROOT), score);
  k_topk<<<1,LDIM,0,stream>>>(score, perm, vals, inv);
  k_nodep<<<CDIV(KSEL*193,256),256,0,stream>>>(node, perm, vals, node_p);
  k_remap<<<CDIV(EDG,256),256,0,stream>>>(PI(IN_ADJ), inv, ns, nd, em);

  // ---- 4x GATv2 ------------------------------------------------------------
  run_gat(stream, node_p, 193, 8, 16, P(IP_GAT0+0),P(IP_GAT0+1),P(IP_GAT0+2),
          P(IP_GAT0+3),P(IP_GAT0+4),P(IP_GAT0+5), ns,nd,em,
          xl,xr,logits,wbuf,mbuf,den, hbuf0, 2);
  run_gat(stream, hbuf0, 128, 8, 16, P(IP_GAT1+0),P(IP_GAT1+1),P(IP_GAT1+2),
          P(IP_GAT1+3),P(IP_GAT1+4),P(IP_GAT1+5), ns,nd,em,
          xl,xr,logits,wbuf,mbuf,den, hbuf1, 2);
  run_gat(stream, hbuf1, 128, 8, 16, P(IP_GAT2+0),P(IP_GAT2+1),P(IP_GAT2+2),
          P(IP_GAT2+3),P(IP_GAT2+4),P(IP_GAT2+5), ns,nd,em,
          xl,xr,logits,wbuf,mbuf,den, hbuf0, 2);
  run_gat(stream, hbuf0, 128, 8,  8, P(IP_GAT3+0),P(IP_GAT3+1),P(IP_GAT3+2),
          P(IP_GAT3+3),P(IP_GAT3+4),P(IP_GAT3+5), ns,nd,em,
          xl,xr,logits,wbuf,mbuf,den, hbuf1, 0);
  k_xnode<<<1,KSEL,0,stream>>>(hbuf1, P(IP_LIN1_W), P(IP_LIN1_B), xnode);

  // ---- voxel scatter + c1d -------------------------------------------------
  k_fillf<<<CDIV(LDIM*VOXD,256),256,0,stream>>>(f3dA, 0.f, (long)LDIM*VOXD);
  k_scatter<<<CDIV(153600,256),256,0,stream>>>(P(IN_VAL), PI(IN_VIDX), f3dA, 153600);
  gemm(stream, f3dA, P(IP_C1D_W), P(IP_C1D_B), nf3d, LDIM, 64, VOXD, 2 /*gelu*/);

  // ---- fused MLP + concat48 + c2d1 ----------------------------------------
  k_fused_cat<<<CDIV(NPIX,128),128,0,stream>>>(P(IN_2D), P(IN_PROFILE), P(IN_ENTROPY),
      P(IN_AFOR), P(IN_MASK3D), P(IP_FUS_W1), P(IP_FUS_B1), P(IP_FUS_W2), P(IP_FUS_B2),
      cat48);
  gemm(stream, cat48, P(IP_C2D1_W), P(IP_C2D1_B), oc, NPIX, 128, 48, 0);

  // ---- instance-norm prep + fused pair GEMM (c2d2) ------------------------
  k_chan_stats<<<64,256,0,stream>>>(nf3d, m1, v1, LDIM, 64);
  k_nf3d_norm<<<CDIV(LDIM*64,256),256,0,stream>>>(nf3d, m1, v1,
      P(IP_IN1_G), P(IP_IN1_B), nf3dN1, nf3dN2);
  k_chan_stats<<<128,256,0,stream>>>(oc, moc, voc, NPIX, 128);
  k_inorm_gelu<<<CDIV((long)NPIX*128,256),256,0,stream>>>(oc, moc, voc,
      P(IP_IN1_G)+128, P(IP_IN1_B)+128, (long)NPIX*128, 128);
  {
    int tiles = (NPIX >> 4) * 4;
    wmma_gemm_pair_f32<<<CDIV(tiles,8),256,0,stream>>>(nf3dN1, nf3dN2, oc,
        P(IP_C2D2_W), P(IP_C2D2_B), pairB);
  }
  k_chan_stats<<<128,256,0,stream>>>(pairB, m2, v2, NPIX, 128);
  k_fillf<<<CDIV((long)PADW*PADW*128,256),256,0,stream>>>(pairP, 0.f, (long)PADW*PADW*128);
  k_inorm_pad<<<CDIV((long)NPIX*128,256),256,0,stream>>>(pairB, m2, v2,
      P(IP_IN2_G), P(IP_IN2_B), pairP);

  // ---- WMMA strided conv 3x3 s2 -> x2 (192x192x128) -----------------------
  k_transw<<<CDIV(9*128*128,256),256,0,stream>>>(P(IP_SAMP_W), Bt);
  {
    int tiles = (NTOK >> 4) * 4;
    wmma_conv_f32<<<CDIV(tiles,8),256,0,stream>>>(pairP, Bt, P(IP_SAMP_B), x2);
  }

  const long nTokEl = (long)NTOK*DMOD;
  // ---- row MHA -------------------------------------------------------------
  k_ln<<<CDIV(NTOK,128),128,0,stream>>>(x2, P(IP_LN1_G), P(IP_LN1_B), xln, NTOK);
  gemm(stream, xln, P(IP_ROW+0), P(IP_ROW+1), qb, NTOK, 128, 128, 0);
  gemm(stream, xln, P(IP_ROW+2), P(IP_ROW+3), kb, NTOK, 128, 128, 0);
  gemm(stream, xln, P(IP_ROW+4), P(IP_ROW+5), vb, NTOK, 128, 128, 0);
  k_attn<<<CDIV(KSEL*4*KSEL,128),128,0,stream>>>(qb, kb, vb, ab, 0);
  gemm(stream, ab, P(IP_ROW+6), P(IP_ROW+7), tmp, NTOK, 128, 128, 0);
  k_add<<<CDIV(nTokEl,256),256,0,stream>>>(x2, tmp, nTokEl);

  // ---- col MHA -------------------------------------------------------------
  k_ln<<<CDIV(NTOK,128),128,0,stream>>>(x2, P(IP_LN2_G), P(IP_LN2_B), xln, NTOK);
  gemm(stream, xln, P(IP_COL+0), P(IP_COL+1), qb, NTOK, 128, 128, 0);
  gemm(stream, xln, P(IP_COL+2), P(IP_COL+3), kb, NTOK, 128, 128, 0);
  gemm(stream, xln, P(IP_COL+4), P(IP_COL+5), vb, NTOK, 128, 128, 0);
  k_attn<<<CDIV(KSEL*4*KSEL,128),128,0,stream>>>(qb, kb, vb, ab, 1);
  gemm(stream, ab, P(IP_COL+6), P(IP_COL+7), tmp, NTOK, 128, 128, 0);
  k_add<<<CDIV(nTokEl,256),256,0,stream>>>(x2, tmp, nTokEl);

  // ---- FFN -----------------------------------------------------------------
  k_ln<<<CDIV(NTOK,128),128,0,stream>>>(x2, P(IP_LN3_G), P(IP_LN3_B), xln, NTOK);
  gemm(stream, xln, P(IP_FF_W1), P(IP_FF_B1), ffh, NTOK, 256, 128, 1 /*relu*/);
  gemm(stream, ffh, P(IP_FF_W2), P(IP_FF_B2), tmp, NTOK, 128, 256, 0);
  k_add<<<CDIV(nTokEl,256),256,0,stream>>>(x2, tmp, nTokEl);

  // ---- mask head + final reduction ----------------------------------------
  k_maskdot<<<CDIV(NTOK,128),128,0,stream>>>(x2, P(IP_MASK_W), P(IP_MASK_B), maskv);
  k_final<<<1,256,0,stream>>>(xnode, maskv, (float*)d_out);
}